// GCN_14602888806986
// MI455X (gfx1250) — compile-verified
//
#include <hip/hip_runtime.h>

typedef float v2f __attribute__((ext_vector_type(2)));
typedef float v8f __attribute__((ext_vector_type(8)));

#define N_FEAT 128   // IN_FEATS == H_FEATS == 128
#define N_CLS  64    // NUM_CLASSES

// ---------------------------------------------------------------------------
// SpMM: out[row] += val * feat[col]   (COO scatter-add, fp32 HW atomics)
// One wave per edge; each lane covers 4 features (32 lanes * 4 = 128).
// ---------------------------------------------------------------------------
__global__ __launch_bounds__(256) void spmm_kernel(
    const int* __restrict__ rows, const int* __restrict__ cols,
    const float* __restrict__ vals, const float* __restrict__ feat,
    float* __restrict__ out, int n_edges)
{
    const int wave = threadIdx.x >> 5;
    const int lane = threadIdx.x & 31;
    const int e = blockIdx.x * 8 + wave;
    if (e >= n_edges) return;

    const int   r = rows[e];
    const int   c = cols[e];
    const float v = vals[e];

    const float4 xv = *(const float4*)(feat + (long)c * N_FEAT + lane * 4);
    float* dst = out + (long)r * N_FEAT + lane * 4;

    __hip_atomic_fetch_add(dst + 0, v * xv.x, __ATOMIC_RELAXED, __HIP_MEMORY_SCOPE_AGENT);
    __hip_atomic_fetch_add(dst + 1, v * xv.y, __ATOMIC_RELAXED, __HIP_MEMORY_SCOPE_AGENT);
    __hip_atomic_fetch_add(dst + 2, v * xv.z, __ATOMIC_RELAXED, __HIP_MEMORY_SCOPE_AGENT);
    __hip_atomic_fetch_add(dst + 3, v * xv.w, __ATOMIC_RELAXED, __HIP_MEMORY_SCOPE_AGENT);
}

// ---------------------------------------------------------------------------
// Dense GEMM tile via V_WMMA_F32_16X16X4_F32 (full fp32 precision).
// One wave -> one 16x16 output tile; K accumulated in steps of 4.
//
// A 16x4 f32 layout (ISA 7.12.2): lanes 0-15 hold M=lane, VGPR0=K0,VGPR1=K1;
//                                 lanes 16-31 hold M=lane-16, VGPR0=K2,VGPR1=K3.
// B 4x16 f32 layout (row-of-N striped across lanes, K split low/high lanes):
//                                 lanes 0-15: VGPRj = row K=j; lanes 16-31: K=2+j.
// C/D 16x16 f32: VGPR r -> M = r (lanes 0-15) or r+8 (lanes 16-31), N = lane%16.
// EXEC stays all-ones through the WMMA loop (stores predicated afterwards).
// ---------------------------------------------------------------------------
template <int NCOLS, bool RELU>
__global__ void wmma_gemm(const float* __restrict__ A,
                          const float* __restrict__ W,
                          float* __restrict__ out, int n_rows)
{
    const int lane = threadIdx.x & 31;
    const int wave = threadIdx.x >> 5;
    const int m0 = blockIdx.x * 16;        // 16-row output tile
    const int n0 = wave * 16;              // 16-col output tile (wave picks cols)
    const int mr = lane & 15;
    const int koff = (lane < 16) ? 0 : 2;

    // Clamp row for loads so EXEC remains full; stores are bounds-checked later.
    int arow = m0 + mr;
    if (arow >= n_rows) arow = n_rows - 1;
    const float* __restrict__ aptr = A + (long)arow * N_FEAT;
    const float* __restrict__ wptr = W + n0 + mr;   // column (n0 + lane%16)

    v8f acc = {};
    #pragma unroll
    for (int kb = 0; kb < N_FEAT; kb += 4) {
        v2f a = *(const v2f*)(aptr + kb + koff);            // A[m][kb+koff .. +1]
        v2f b;
        b.x = wptr[(kb + koff + 0) * NCOLS];                // B[kb+koff  ][n]
        b.y = wptr[(kb + koff + 1) * NCOLS];                // B[kb+koff+1][n]
        acc = __builtin_amdgcn_wmma_f32_16x16x4_f32(
            /*neg_a=*/false, a, /*neg_b=*/false, b,
            /*c_mod=*/(short)0, acc, /*reuse_a=*/false, /*reuse_b=*/false);
    }

    const int nc = n0 + mr;
    const int mbase = m0 + ((lane < 16) ? 0 : 8);
    #pragma unroll
    for (int r = 0; r < 8; ++r) {
        const int m = mbase + r;
        if (m < n_rows) {
            float v = acc[r];
            if (RELU) v = fmaxf(v, 0.0f);
            out[(long)m * NCOLS + nc] = v;
        }
    }
}

// ---------------------------------------------------------------------------
// Orchestration:  agg = spmm(x); h = relu(agg @ W1); agg = spmm(h);
//                 out = agg @ W2
// ---------------------------------------------------------------------------
extern "C" void kernel_launch(void* const* d_in, const int* in_sizes, int n_in,
                              void* d_out, int out_size, void* d_ws, size_t ws_size,
                              hipStream_t stream)
{
    const float* x  = (const float*)d_in[0];
    const int*   ei = (const int*)  d_in[1];
    const float* ev = (const float*)d_in[2];
    const float* W1 = (const float*)d_in[3];
    const float* W2 = (const float*)d_in[4];
    float* out = (float*)d_out;

    const int n_nodes = in_sizes[0] / N_FEAT;   // 100000
    const int n_edges = in_sizes[2];            // 1600000
    const int* rows = ei;                       // edge_index[0]
    const int* cols = ei + n_edges;             // edge_index[1]

    const size_t agg_bytes = (size_t)n_nodes * N_FEAT * sizeof(float);
    float* agg = (float*)d_ws;
    float* h   = (float*)((char*)d_ws + agg_bytes);

    const int spmm_blocks = (n_edges + 7) / 8;          // 8 waves (edges) / block
    const int mtiles      = (n_nodes + 15) / 16;

    // Layer 1
    hipMemsetAsync(agg, 0, agg_bytes, stream);
    spmm_kernel<<<spmm_blocks, 256, 0, stream>>>(rows, cols, ev, x, agg, n_edges);
    wmma_gemm<N_FEAT, true><<<mtiles, 256, 0, stream>>>(agg, W1, h, n_nodes);   // 8 waves = 128 cols

    // Layer 2
    hipMemsetAsync(agg, 0, agg_bytes, stream);
    spmm_kernel<<<spmm_blocks, 256, 0, stream>>>(rows, cols, ev, h, agg, n_edges);
    wmma_gemm<N_CLS, false><<<mtiles, 128, 0, stream>>>(agg, W2, out, n_nodes); // 4 waves = 64 cols
}